// GAT_PYG_58110907515579
// MI455X (gfx1250) — compile-verified
//
#include <hip/hip_runtime.h>
#include <hip/hip_bf16.h>

// ---------------------------------------------------------------------------
// GAT (2-layer, PyG semantics) for MI455X / gfx1250.
//  - GEMMs: V_WMMA_F32_16X16X4_F32, 256-thread blocks = 8 waves x 8 M-tiles,
//    B panel (K x 16) staged once in LDS and reused by all waves.
//  - segment softmax via ordered-uint atomicMax + recompute passes
//  - edge gather/scatter vectorized to b128; atomics f32 (L2-resident, ~80MB)
// ---------------------------------------------------------------------------

typedef __attribute__((ext_vector_type(2))) float v2f;
typedef __attribute__((ext_vector_type(8))) float v8f;

#define NEG_SLOPE 0.2f

// ---- float ordering tricks for atomic segment-max --------------------------
__device__ __forceinline__ unsigned f32_key(float f) {
    unsigned u = __float_as_uint(f);
    return (u & 0x80000000u) ? ~u : (u | 0x80000000u);
}
__device__ __forceinline__ float f32_unkey(unsigned u) {
    unsigned b = (u & 0x80000000u) ? (u & 0x7FFFFFFFu) : ~u;
    return __uint_as_float(b);
}
__device__ __forceinline__ float leaky(float x) {
    return x > 0.0f ? x : NEG_SLOPE * x;
}
__device__ __forceinline__ void edge_pair(const int* __restrict__ ei, int E,
                                          int e, int& s, int& d) {
    if (e < E) { s = ei[e]; d = ei[E + e]; }
    else       { s = e - E; d = s; }          // self loops appended
}

// ---------------------------------------------------------------------------
// f32 WMMA GEMM: C[M x N] = A[M x K] @ B[K x N], row-major. K <= 128.
// One 256-thread block = 8 waves; each wave owns one 16x16 C tile in the same
// column stripe; the K x 16 B panel lives in LDS (loaded once per block).
// Fragment layouts per CDNA5 ISA 7.12.2:
//   A 16x4 : lanes 0-15 rows M=0..15, VGPR0=K0 VGPR1=K1; lanes 16-31 K2/K3
//   B 4x16 : VGPR0 row K0 (lanes 0-15) / K2 (lanes 16-31); VGPR1 K1/K3
//   C 16x16: VGPR r -> row r (lanes 0-15) / row r+8 (lanes 16-31)
// ---------------------------------------------------------------------------
#define GEMM_WAVES 8
__global__ __launch_bounds__(256)
void gemm_wmma_f32(const float* __restrict__ A, const float* __restrict__ B,
                   float* __restrict__ C, int M, int N, int K) {
    __shared__ float Bs[128 * 16];            // max K = 128

    const int tid  = threadIdx.x;
    const int col0 = blockIdx.y * 16;

    // cooperative load of the B panel [K][16] into LDS
    for (int i = tid; i < K * 16; i += 256) {
        const int kk = i >> 4, cc = i & 15;
        Bs[i] = B[(size_t)kk * N + (col0 + cc)];
    }
    __syncthreads();

    const int wave = tid >> 5;
    const int lane = tid & 31;
    const int half = lane >> 4;               // 0 | 1
    const int l16  = lane & 15;
    const int row0 = (blockIdx.x * GEMM_WAVES + wave) * 16;
    if (row0 >= M) return;                    // no further barriers below

    const float* ap = A + (size_t)(row0 + l16) * K + half * 2;  // step +4
    const float* bs = Bs + (half * 2) * 16 + l16;               // step +64

    v8f acc = {};
    for (int k = 0; k < K; k += 4) {
        const v2f a = *(const v2f*)ap;        // global_load_b64
        v2f b; b.x = bs[0]; b.y = bs[16];     // ds_load x2
        ap += 4;
        bs += 64;
        acc = __builtin_amdgcn_wmma_f32_16x16x4_f32(
            /*neg_a=*/false, a, /*neg_b=*/false, b,
            /*c_mod=*/(short)0, acc, /*reuse_a=*/false, /*reuse_b=*/false);
    }
#pragma unroll
    for (int r = 0; r < 8; ++r) {
        const int row = row0 + r + half * 8;
        C[(size_t)row * N + (col0 + l16)] = acc[r];
    }
}

// ---------------------------------------------------------------------------
// Per-(node,head): attention dot products + (re)initialize all accumulators.
// h layout: [N][H][C] row-major; thread i = n*H + hh. C in {8,16} (mult of 4).
// ---------------------------------------------------------------------------
__global__ void attn_prep(const float* __restrict__ h,
                          const float* __restrict__ a_src,
                          const float* __restrict__ a_dst,
                          float* __restrict__ al_s, float* __restrict__ al_d,
                          unsigned* __restrict__ mkey, float* __restrict__ denom,
                          float* __restrict__ outacc,
                          int n_nodes, int H, int C) {
    const int i = blockIdx.x * blockDim.x + threadIdx.x;
    if (i >= n_nodes * H) return;
    const int hh = i % H;
    const float4* hp = (const float4*)(h + (size_t)i * C);
    const float4* as = (const float4*)(a_src + hh * C);
    const float4* ad = (const float4*)(a_dst + hh * C);
    float s = 0.f, d = 0.f;
    for (int c = 0; c < (C >> 2); ++c) {
        const float4 hv = hp[c], av = as[c], dv = ad[c];
        s += hv.x * av.x + hv.y * av.y + hv.z * av.z + hv.w * av.w;
        d += hv.x * dv.x + hv.y * dv.y + hv.z * dv.z + hv.w * dv.w;
    }
    al_s[i] = s;
    al_d[i] = d;
    mkey[i] = f32_key(-INFINITY);
    denom[i] = 0.f;
    float4* op = (float4*)(outacc + (size_t)i * C);
    const float4 z = {0.f, 0.f, 0.f, 0.f};
    for (int c = 0; c < (C >> 2); ++c) op[c] = z;
}

// ---- pass 1: segment max of leaky-relu logits over destination -------------
__global__ void edge_max(const int* __restrict__ ei, int E, int n_nodes, int H,
                         const float* __restrict__ al_s,
                         const float* __restrict__ al_d,
                         unsigned* __restrict__ mkey) {
    const int t = blockIdx.x * blockDim.x + threadIdx.x;
    const int total = (E + n_nodes) * H;
    if (t >= total) return;
    const int e = t / H, hh = t - e * H;
    int s, d; edge_pair(ei, E, e, s, d);
    const float lg = leaky(al_s[(size_t)s * H + hh] + al_d[(size_t)d * H + hh]);
    atomicMax(&mkey[(size_t)d * H + hh], f32_key(lg));
}

// ---- pass 2: segment sum of exp(logit - max) -------------------------------
__global__ void edge_expsum(const int* __restrict__ ei, int E, int n_nodes, int H,
                            const float* __restrict__ al_s,
                            const float* __restrict__ al_d,
                            const unsigned* __restrict__ mkey,
                            float* __restrict__ denom) {
    const int t = blockIdx.x * blockDim.x + threadIdx.x;
    const int total = (E + n_nodes) * H;
    if (t >= total) return;
    const int e = t / H, hh = t - e * H;
    int s, d; edge_pair(ei, E, e, s, d);
    const size_t di = (size_t)d * H + hh;
    const float lg = leaky(al_s[(size_t)s * H + hh] + al_d[di]);
    atomicAdd(&denom[di], __expf(lg - f32_unkey(mkey[di])));
}

// ---- pass 3: normalized weighted scatter of source features ----------------
__global__ void edge_scatter(const int* __restrict__ ei, int E, int n_nodes,
                             int H, int C,
                             const float* __restrict__ al_s,
                             const float* __restrict__ al_d,
                             const unsigned* __restrict__ mkey,
                             const float* __restrict__ denom,
                             const float* __restrict__ h,
                             float* __restrict__ outacc) {
    const int t = blockIdx.x * blockDim.x + threadIdx.x;
    const int total = (E + n_nodes) * H;
    if (t >= total) return;
    const int e = t / H, hh = t - e * H;
    int s, d; edge_pair(ei, E, e, s, d);
    const size_t di = (size_t)d * H + hh;
    const float lg   = leaky(al_s[(size_t)s * H + hh] + al_d[di]);
    const float w    = __expf(lg - f32_unkey(mkey[di]));
    const float coef = w / denom[di];
    const float4* hp = (const float4*)(h + ((size_t)s * H + hh) * C);   // b128 gather
    float* op = outacc + di * C;
    for (int c4 = 0; c4 < (C >> 2); ++c4) {
        const float4 hv = hp[c4];
        float* o = op + c4 * 4;
        atomicAdd(&o[0], hv.x * coef);
        atomicAdd(&o[1], hv.y * coef);
        atomicAdd(&o[2], hv.z * coef);
        atomicAdd(&o[3], hv.w * coef);
    }
}

// ---- bias + ReLU, in place (n multiple of 4, F multiple of 4) --------------
__global__ void bias_relu_inplace(float* __restrict__ x,
                                  const float* __restrict__ b, int n, int F) {
    const int i4 = blockIdx.x * blockDim.x + threadIdx.x;
    if (i4 >= (n >> 2)) return;
    const int base = i4 << 2;
    float4 v = ((const float4*)x)[i4];
    const float4 bv = *(const float4*)(b + (base % F));
    v.x = fmaxf(v.x + bv.x, 0.f);
    v.y = fmaxf(v.y + bv.y, 0.f);
    v.z = fmaxf(v.z + bv.z, 0.f);
    v.w = fmaxf(v.w + bv.w, 0.f);
    ((float4*)x)[i4] = v;
}

// ---- bias + log_softmax over C=16 classes ----------------------------------
__global__ void finalize_logsoftmax(const float* __restrict__ acc,
                                    const float* __restrict__ bias,
                                    float* __restrict__ out,
                                    int n_nodes, int C) {
    const int n = blockIdx.x * blockDim.x + threadIdx.x;
    if (n >= n_nodes) return;
    float v[16];
    const float4* ap = (const float4*)(acc + (size_t)n * C);
    const float4* bp = (const float4*)bias;
    float m = -INFINITY;
    for (int c4 = 0; c4 < (C >> 2); ++c4) {
        const float4 a = ap[c4], b = bp[c4];
        v[c4 * 4 + 0] = a.x + b.x;
        v[c4 * 4 + 1] = a.y + b.y;
        v[c4 * 4 + 2] = a.z + b.z;
        v[c4 * 4 + 3] = a.w + b.w;
    }
    for (int c = 0; c < C; ++c) m = fmaxf(m, v[c]);
    float s = 0.f;
    for (int c = 0; c < C; ++c) s += __expf(v[c] - m);
    const float ls = __logf(s) + m;
    float4* op = (float4*)(out + (size_t)n * C);
    for (int c4 = 0; c4 < (C >> 2); ++c4) {
        float4 o;
        o.x = v[c4 * 4 + 0] - ls;
        o.y = v[c4 * 4 + 1] - ls;
        o.z = v[c4 * 4 + 2] - ls;
        o.w = v[c4 * 4 + 3] - ls;
        op[c4] = o;
    }
}

// ---------------------------------------------------------------------------
extern "C" void kernel_launch(void* const* d_in, const int* in_sizes, int n_in,
                              void* d_out, int out_size, void* d_ws, size_t ws_size,
                              hipStream_t stream) {
    const float* x   = (const float*)d_in[0];
    const int*   ei  = (const int*)  d_in[1];
    const float* W1  = (const float*)d_in[2];
    const float* as1 = (const float*)d_in[3];
    const float* ad1 = (const float*)d_in[4];
    const float* b1  = (const float*)d_in[5];
    const float* W2  = (const float*)d_in[6];
    const float* as2 = (const float*)d_in[7];
    const float* ad2 = (const float*)d_in[8];
    const float* b2  = (const float*)d_in[9];

    const int F_IN = 128, H1 = 8, C1 = 8, F1 = H1 * C1 /*64*/, C2 = 16;
    const int N = in_sizes[0] / F_IN;       // 100000
    const int E = in_sizes[1] / 2;          // 1.6M

    // ---- workspace layout (floats) ----
    float* ws = (float*)d_ws;
    size_t off = 0;
    float*    h1   = ws + off; off += (size_t)N * F1;   // layer1 features
    float*    out1 = ws + off; off += (size_t)N * F1;   // layer1 accum -> relu'd h
    float*    als1 = ws + off; off += (size_t)N * H1;
    float*    ald1 = ws + off; off += (size_t)N * H1;
    unsigned* m1   = (unsigned*)(ws + off); off += (size_t)N * H1;
    float*    den1 = ws + off; off += (size_t)N * H1;
    float*    h2   = ws + off; off += (size_t)N * C2;
    float*    out2 = ws + off; off += (size_t)N * C2;
    float*    als2 = ws + off; off += (size_t)N;
    float*    ald2 = ws + off; off += (size_t)N;
    unsigned* m2   = (unsigned*)(ws + off); off += (size_t)N;
    float*    den2 = ws + off; off += (size_t)N;

    const int TB = 256;
    const int mtiles = N / 16;                               // 6250
    const int mblocks = (mtiles + GEMM_WAVES - 1) / GEMM_WAVES;

    // ================= layer 1 =================
    gemm_wmma_f32<<<dim3(mblocks, F1 / 16), TB, 0, stream>>>(x, W1, h1, N, F1, F_IN);

    const int nh1 = N * H1;
    attn_prep<<<(nh1 + TB - 1) / TB, TB, 0, stream>>>(h1, as1, ad1, als1, ald1,
                                                      m1, den1, out1, N, H1, C1);
    const int et1 = (E + N) * H1;
    edge_max    <<<(et1 + TB - 1) / TB, TB, 0, stream>>>(ei, E, N, H1, als1, ald1, m1);
    edge_expsum <<<(et1 + TB - 1) / TB, TB, 0, stream>>>(ei, E, N, H1, als1, ald1, m1, den1);
    edge_scatter<<<(et1 + TB - 1) / TB, TB, 0, stream>>>(ei, E, N, H1, C1, als1, ald1,
                                                         m1, den1, h1, out1);
    bias_relu_inplace<<<((N * F1 / 4) + TB - 1) / TB, TB, 0, stream>>>(out1, b1, N * F1, F1);

    // ================= layer 2 (H=1, concat=False -> mean of 1 head) =======
    gemm_wmma_f32<<<dim3(mblocks, C2 / 16), TB, 0, stream>>>(out1, W2, h2, N, C2, F1);

    attn_prep<<<(N + TB - 1) / TB, TB, 0, stream>>>(h2, as2, ad2, als2, ald2,
                                                    m2, den2, out2, N, 1, C2);
    const int et2 = E + N;
    edge_max    <<<(et2 + TB - 1) / TB, TB, 0, stream>>>(ei, E, N, 1, als2, ald2, m2);
    edge_expsum <<<(et2 + TB - 1) / TB, TB, 0, stream>>>(ei, E, N, 1, als2, ald2, m2, den2);
    edge_scatter<<<(et2 + TB - 1) / TB, TB, 0, stream>>>(ei, E, N, 1, C2, als2, ald2,
                                                         m2, den2, h2, out2);

    finalize_logsoftmax<<<(N + TB - 1) / TB, TB, 0, stream>>>(out2, b2, (float*)d_out, N, C2);
}